// SegmentCSR_38843684225660
// MI455X (gfx1250) — compile-verified
//
#include <hip/hip_runtime.h>

typedef float v2f __attribute__((ext_vector_type(2)));
typedef float v8f __attribute__((ext_vector_type(8)));

#define FEAT 128
#define WAVES_PER_BLOCK 8

// One wave per segment. Segment sum over rows [indptr[s], indptr[s+1]) of
// x[*][128] via V_WMMA_F32_16X16X4_F32 with A == ones:
//   D[m][n] = sum_{k=0..3} B[k][n] + C[m][n]
// B (4x16 fp32, 2 VGPRs): lanes 0-15 hold K=0 (VGPR0) / K=1 (VGPR1) at
// N=lane; lanes 16-31 hold K=2 / K=3 at N=lane-16. Eight B blocks cover the
// 128 columns; all 16 D rows carry identical column sums.
// Segment bounds are scalarized (readfirstlane) so loop control is pure SALU
// and EXEC stays all-ones around the WMMAs; the ragged tail uses clamped
// in-bounds loads + v_cndmask zeroing instead of exec-masked loads.
__global__ __launch_bounds__(256)
void segcsr_wmma_kernel(const float* __restrict__ x,
                        const long long* __restrict__ indptr,
                        float* __restrict__ out,
                        int n_seg)
{
    const int wave = threadIdx.x >> 5;
    const int lane = threadIdx.x & 31;
    const int seg  = blockIdx.x * WAVES_PER_BLOCK + wave;
    if (seg >= n_seg) return;

    // Wave-uniform bounds; force them into SGPRs (rows fit in 32 bits).
    int start = (int)indptr[seg];
    int end   = (int)indptr[seg + 1];
    start = __builtin_amdgcn_readfirstlane(start);
    end   = __builtin_amdgcn_readfirstlane(end);
    const int len   = end - start;
    const int nfull = len & ~3;   // rows handled by unpredicated chunks
    const int rem   = len & 3;    // ragged tail rows

    const int half = lane >> 4;   // 0 -> holds K=0,1 ; 1 -> holds K=2,3
    const int col  = lane & 15;   // N within a 16-column block

    v8f acc[8];
    #pragma unroll
    for (int j = 0; j < 8; ++j) acc[j] = (v8f){0,0,0,0,0,0,0,0};

    const v2f ones = {1.0f, 1.0f};

    // Per-lane base: row (start + 2*half), column col. Row stride folds into
    // load immediates; advance by 4 rows (2 KB) per chunk.
    const float* p = x + (start + 2 * half) * FEAT + col;

    for (int i = 0; i < nfull; i += 4) {
        v2f b[8];
        #pragma unroll
        for (int j = 0; j < 8; ++j) {
            b[j].x = __builtin_nontemporal_load(p + 16 * j);         // K = 2h
            b[j].y = __builtin_nontemporal_load(p + FEAT + 16 * j);  // K = 2h+1
        }
        #pragma unroll
        for (int j = 0; j < 8; ++j) {
            acc[j] = __builtin_amdgcn_wmma_f32_16x16x4_f32(
                /*neg_a=*/false, ones, /*neg_b=*/false, b[j],
                /*c_mod=*/(short)0, acc[j],
                /*reuse_a=*/false, /*reuse_b=*/false);
        }
        p += 4 * FEAT;
    }

    if (rem) {  // scalar branch: rem is uniform
        const int k0 = 2 * half;
        const int k1 = k0 + 1;
        const bool in0 = (k0 < rem);
        const bool in1 = (k1 < rem);
        // Clamp to last valid tail row -> loads always in-bounds; zero the
        // out-of-range K rows with a select (no exec masking).
        const int r0 = nfull + (in0 ? k0 : rem - 1);
        const int r1 = nfull + (in1 ? k1 : rem - 1);
        const float* q0 = x + (start + r0) * FEAT + col;
        const float* q1 = x + (start + r1) * FEAT + col;

        v2f b[8];
        #pragma unroll
        for (int j = 0; j < 8; ++j) {
            const float lo = __builtin_nontemporal_load(q0 + 16 * j);
            const float hi = __builtin_nontemporal_load(q1 + 16 * j);
            b[j].x = in0 ? lo : 0.0f;
            b[j].y = in1 ? hi : 0.0f;
        }
        #pragma unroll
        for (int j = 0; j < 8; ++j) {
            acc[j] = __builtin_amdgcn_wmma_f32_16x16x4_f32(
                false, ones, false, b[j], (short)0, acc[j], false, false);
        }
    }

    // Every lane's acc element 0 holds its column's total (all M rows equal).
    // Lanes 0-15 write column blocks 0-3, lanes 16-31 write blocks 4-7:
    // two coalesced 64 B runs per store.
    float* o = out + (size_t)seg * FEAT + half * 64 + col;
    #pragma unroll
    for (int jj = 0; jj < 4; ++jj) {
        const float v0 = acc[jj][0];      // block jj     (half == 0)
        const float v1 = acc[4 + jj][0];  // block 4 + jj (half == 1)
        __builtin_nontemporal_store(half ? v1 : v0, o + 16 * jj);
    }
}

extern "C" void kernel_launch(void* const* d_in, const int* in_sizes, int n_in,
                              void* d_out, int out_size, void* d_ws, size_t ws_size,
                              hipStream_t stream) {
    const float*     x      = (const float*)d_in[0];
    const long long* indptr = (const long long*)d_in[1];
    float*           out    = (float*)d_out;

    const int n_seg  = in_sizes[1] - 1;  // indptr has N_SEG + 1 entries
    const int blocks = (n_seg + WAVES_PER_BLOCK - 1) / WAVES_PER_BLOCK;
    segcsr_wmma_kernel<<<blocks, 256, 0, stream>>>(x, indptr, out, n_seg);
}